// VNAttentionWithContext2_78254304133834
// MI455X (gfx1250) — compile-verified
//
#include <hip/hip_runtime.h>
#include <hip/hip_bf16.h>
#include <math.h>

// Problem constants (match reference)
#define B_ 32
#define D_ 1024
#define V_ 3
#define L_ 256
#define N_ (B_ * V_) // 96 batched (b,v) columns

typedef float v2f __attribute__((ext_vector_type(2)));
typedef float v8f __attribute__((ext_vector_type(8)));

__device__ __forceinline__ float wave_sum(float v) {
#pragma unroll
  for (int o = 16; o > 0; o >>= 1) v += __shfl_xor(v, o, 32);
  return v;
}
__device__ __forceinline__ float wave_max(float v) {
#pragma unroll
  for (int o = 16; o > 0; o >>= 1) v = fmaxf(v, __shfl_xor(v, o, 32));
  return v;
}

// ---------------------------------------------------------------------------
// Kernel 1: XWt[d][n] = sum_l x[b,d,v,l] * W3[l],  n = b*V + v
// One wave per (b,d,v) job; job index == flat (b,d,v) so the 256-float row of
// x is contiguous and the wave load is fully coalesced.
// ---------------------------------------------------------------------------
__global__ void k_xw(const float* __restrict__ x, const float* __restrict__ W3,
                     float* __restrict__ XWt) {
  int job  = blockIdx.x * 8 + (threadIdx.x >> 5); // 8 waves / block
  int lane = threadIdx.x & 31;
  const float* xr = x + (size_t)job * L_;
  float acc = 0.f;
#pragma unroll
  for (int j = 0; j < 8; ++j) {
    int l = lane + 32 * j;
    acc = fmaf(xr[l], W3[l], acc);
  }
  acc = wave_sum(acc);
  if (lane == 0) {
    int v  = job % V_;
    int bd = job / V_;
    int d  = bd % D_;
    int b  = bd / D_;
    XWt[d * N_ + (b * V_ + v)] = acc;
  }
}

// ---------------------------------------------------------------------------
// Kernel 2/3: C(D x 96) = op(A)(D x D) @ Bm(D x 96) via V_WMMA_F32_16X16X4_F32
// TRANSA=0: A row-major (W2).  TRANSA=1: use A^T (W1^T).
// One wave per 16x16 output tile; K-loop over D in steps of 4.
// Grid: (D/16)*(96/16) = 384 waves = 96 blocks x 128 threads (all waves full,
// EXEC all-ones as WMMA requires).
// ---------------------------------------------------------------------------
template <int TRANSA>
__global__ void k_gemm(const float* __restrict__ A, const float* __restrict__ Bm,
                       float* __restrict__ C) {
  int gw   = blockIdx.x * 4 + (threadIdx.x >> 5); // 0..383
  int lane = threadIdx.x & 31;
  const int NTN = N_ / 16; // 6
  int tm = gw / NTN;       // 0..63
  int tn = gw % NTN;       // 0..5
  int m0 = tm * 16, n0 = tn * 16;

  int ml = lane & 15;        // row within A tile / col within B tile
  int kh = (lane >> 4) * 2;  // K sub-offset: lanes 0-15 -> K{0,1}, 16-31 -> K{2,3}

  v8f c = {};
  for (int k0 = 0; k0 < D_; k0 += 4) {
    int ka = k0 + kh;
    v2f a, b;
    int m = m0 + ml;
    if (TRANSA) { // A(m,k) = A[k*D + m]
      a.x = A[(size_t)ka * D_ + m];
      a.y = A[(size_t)(ka + 1) * D_ + m];
    } else {      // A(m,k) = A[m*D + k]
      a.x = A[(size_t)m * D_ + ka];
      a.y = A[(size_t)m * D_ + ka + 1];
    }
    int n = n0 + ml;
    b.x = Bm[ka * N_ + n];
    b.y = Bm[(ka + 1) * N_ + n];
    // 8 args: (neg_a, A, neg_b, B, c_mod, C, reuse_a, reuse_b)
    c = __builtin_amdgcn_wmma_f32_16x16x4_f32(false, a, false, b, (short)0, c,
                                              false, false);
  }
  // C/D layout: VGPR r -> row m0 + r + 8*(lane>>4), col n0 + (lane&15)
  int nn    = n0 + ml;
  int mbase = m0 + ((lane >> 4) << 3);
#pragma unroll
  for (int r = 0; r < 8; ++r) C[(size_t)(mbase + r) * N_ + nn] = c[r];
}

// ---------------------------------------------------------------------------
// Kernel 4: per (b,v) block (256 threads):
//   ait[l] = sum_d G[d][n] * x[b,d,v,l]   (thread-per-l, coalesced over l)
//   attn   = softmax_l(ait)
//   out[b,d,v] = sum_l attn[l] * x[b,d,v,l]  (wave-per-d, attn from LDS)
// ---------------------------------------------------------------------------
__global__ void k_attn_out(const float* __restrict__ x, const float* __restrict__ G,
                           float* __restrict__ out) {
  __shared__ float attn[L_];
  __shared__ float red[8];

  int bv = blockIdx.x;       // 0..95
  int b = bv / V_, v = bv % V_;
  int t = threadIdx.x;       // 0..255 (== l in phase 1)
  int wid = t >> 5, lane = t & 31;

  size_t xbase = ((size_t)b * D_ * V_ + v) * L_; // &x[b, d=0, v, l=0]

  // ---- phase 1: ait for l = t -------------------------------------------
  float acc = 0.f;
#pragma unroll 4
  for (int d = 0; d < D_; ++d) {
    const float* xr = x + xbase + (size_t)d * (V_ * L_);
    __builtin_prefetch(xr + V_ * L_, 0, 1); // global_prefetch of next row
    acc = fmaf(G[d * N_ + bv], xr[t], acc);
  }

  // ---- softmax over the 256 l's -----------------------------------------
  float m = wave_max(acc);
  if (lane == 0) red[wid] = m;
  __syncthreads();
  float bmax = red[0];
#pragma unroll
  for (int i = 1; i < 8; ++i) bmax = fmaxf(bmax, red[i]);
  float e = expf(acc - bmax);
  __syncthreads(); // everyone done reading red[] before reuse
  float s = wave_sum(e);
  if (lane == 0) red[wid] = s;
  __syncthreads();
  float bsum = red[0];
#pragma unroll
  for (int i = 1; i < 8; ++i) bsum += red[i];
  attn[t] = e / bsum;
  __syncthreads();

  // ---- phase 2: out[b,d,v] = sum_l attn[l] * x[b,d,v,l] ------------------
  for (int d = wid; d < D_; d += 8) {
    const float* xr = x + xbase + (size_t)d * (V_ * L_);
    float ssum = 0.f;
#pragma unroll
    for (int j = 0; j < 8; ++j) {
      int l = lane + 32 * j;
      ssum = fmaf(xr[l], attn[l], ssum);
    }
    ssum = wave_sum(ssum);
    if (lane == 0) out[((size_t)b * D_ + d) * V_ + v] = ssum;
  }
}

// ---------------------------------------------------------------------------
extern "C" void kernel_launch(void* const* d_in, const int* in_sizes, int n_in,
                              void* d_out, int out_size, void* d_ws, size_t ws_size,
                              hipStream_t stream) {
  const float* x  = (const float*)d_in[0]; // (B,D,V,L)
  const float* W1 = (const float*)d_in[1]; // (D,D)
  const float* W2 = (const float*)d_in[2]; // (D,D)
  const float* W3 = (const float*)d_in[3]; // (1,L)
  float* out = (float*)d_out;              // (B,D,V)

  float* ws  = (float*)d_ws;
  float* XWt = ws;                 // D x 96
  float* HSW = ws + D_ * N_;       // D x 96
  float* G   = ws + 2 * D_ * N_;   // D x 96

  // 1) XWt = reduce_l( x * W3 )  : B*D*V waves, 8 waves/block
  {
    int jobs = B_ * D_ * V_;        // 98304
    int blocks = jobs / 8;          // 12288
    k_xw<<<blocks, 256, 0, stream>>>(x, W3, XWt);
  }
  // 2) HSW = W2 @ XWt   (fp32 WMMA)
  k_gemm<0><<<96, 128, 0, stream>>>(W2, XWt, HSW);
  // 3) G = W1^T @ HSW   (fp32 WMMA)
  k_gemm<1><<<96, 128, 0, stream>>>(W1, HSW, G);
  // 4) ait -> softmax -> out
  k_attn_out<<<N_, 256, 0, stream>>>(x, G, out);
}